// FullModelWithCanvasPainter_70325794505068
// MI455X (gfx1250) — compile-verified
//
#include <hip/hip_runtime.h>
#include <hip/hip_bf16.h>
#include <math.h>

// ---------------------------------------------------------------------------
// Dims (fixed by the reference)
// ---------------------------------------------------------------------------
#define BB 64
#define NN 64
#define DMODEL 512
#define DSTATE 512
#define DATT 512
#define HH 128
#define WW 128
#define CC 16
#define KK 5
#define TT 32
#define HALF 2
#define HP (HH + 2 * HALF)   // 132
#define WP (WW + 2 * HALF)   // 132

typedef __attribute__((ext_vector_type(2))) float v2f;
typedef __attribute__((ext_vector_type(8))) float v8f;

__device__ __forceinline__ v8f wmma_f32_4(v2f a, v2f b, v8f c) {
    // V_WMMA_F32_16X16X4_F32 : D(16x16 f32) = A(16x4 f32) x B(4x16 f32) + C
    return __builtin_amdgcn_wmma_f32_16x16x4_f32(
        /*neg_a=*/false, a, /*neg_b=*/false, b,
        /*c_mod=*/(short)0, c, /*reuse_a=*/false, /*reuse_b=*/false);
}

__device__ __forceinline__ float sigmoidf_(float x) {
    return 1.0f / (1.0f + expf(-x));
}

__device__ __forceinline__ v2f ld2(const float* p) {
    return *(const v2f*)p;   // 8B-aligned by construction -> global_load_b64
}

// ---------------------------------------------------------------------------
// Generic fp32 WMMA GEMM:  out[M,N] = [A1 | A2] @ W^T + bias, optional tanh.
// A1: [M,K1], A2: [M,K2] (may be null, K2=0), W: [N, K1+K2], all row-major.
// One wave32 computes one 16-row M tile x NB consecutive 16-col N tiles.
// Each k-chunk: 1 A load feeds NB WMMAs (independent accumulator chains).
// K1, K2 multiples of 4; addresses 8B aligned.
// ---------------------------------------------------------------------------
template <int NB>
__global__ void __launch_bounds__(32)
wmma_gemm_xt(const float* __restrict__ A1, int K1,
             const float* __restrict__ A2, int K2,
             const float* __restrict__ W, const float* __restrict__ bias,
             float* __restrict__ out, int N, int act, int ngroupsN)
{
    const int tile = blockIdx.x;
    const int mt = tile / ngroupsN;
    const int gt = tile % ngroupsN;
    const int lane = threadIdx.x & 31;
    const int half = lane >> 4;      // 0: holds K{0,1}; 1: holds K{2,3}
    const int lr   = lane & 15;
    const int K = K1 + K2;

    const int arow = mt * 16 + lr;

    v8f acc[NB];
    const float* wp[NB];
    int bcol[NB];
#pragma unroll
    for (int j = 0; j < NB; ++j) {
        acc[j] = (v8f){0.f, 0.f, 0.f, 0.f, 0.f, 0.f, 0.f, 0.f};
        bcol[j] = (gt * NB + j) * 16 + lr;
        wp[j] = W + (size_t)bcol[j] * K + 2 * half;
        __builtin_prefetch(wp[j], 0, 0);            // global_prefetch_b8 (L2-resident weights)
        __builtin_prefetch(wp[j] + 128, 0, 0);
    }

    // ---- phase 1: K over A1 ----
    {
        const float* ap = A1 + (size_t)arow * K1 + 2 * half;
        for (int k0 = 0; k0 < K1; k0 += 4) {
            const v2f a = ld2(ap); ap += 4;
#pragma unroll
            for (int j = 0; j < NB; ++j) {
                const v2f b = ld2(wp[j]); wp[j] += 4;
                acc[j] = wmma_f32_4(a, b, acc[j]);
            }
        }
    }
    // ---- phase 2: K over A2 (w pointers continue through W's K axis) ----
    if (K2 > 0) {
        const float* ap = A2 + (size_t)arow * K2 + 2 * half;
        for (int k0 = 0; k0 < K2; k0 += 4) {
            const v2f a = ld2(ap); ap += 4;
#pragma unroll
            for (int j = 0; j < NB; ++j) {
                const v2f b = ld2(wp[j]); wp[j] += 4;
                acc[j] = wmma_f32_4(a, b, acc[j]);
            }
        }
    }

#pragma unroll
    for (int j = 0; j < NB; ++j) {
        const float bv = bias ? bias[bcol[j]] : 0.0f;
#pragma unroll
        for (int i = 0; i < 8; ++i) {
            const int row = mt * 16 + i + 8 * half;  // C/D layout: VGPR i -> M=i (+8 hi half)
            float v = acc[j][i] + bv;
            if (act == 1) v = tanhf(v);
            out[(size_t)row * N + bcol[j]] = v;
        }
    }
}

// ---------------------------------------------------------------------------
// Fused GRU step: s_new = GRU(x=[r|g], s), torch gate order (r,z,n).
// w_ih: [1536,1024], w_hh: [1536,512]. One wave per (16-row, 16-col) slice;
// 6 WMMA accumulator tiles (ir,iz,in / hr,hz,hn), gate math in registers.
// Each A load feeds 3 WMMAs.
// ---------------------------------------------------------------------------
__global__ void __launch_bounds__(32)
gru_kernel(const float* __restrict__ r, const float* __restrict__ g,
           const float* __restrict__ s,
           const float* __restrict__ w_ih, const float* __restrict__ b_ih,
           const float* __restrict__ w_hh, const float* __restrict__ b_hh,
           float* __restrict__ s_out)
{
    const int mt = blockIdx.x >> 5;   // 0..3  (64 rows / 16)
    const int jt = blockIdx.x & 31;   // 0..31 (512 cols / 16)
    const int lane = threadIdx.x & 31;
    const int half = lane >> 4;
    const int lr   = lane & 15;

    const int arow = mt * 16 + lr;
    const int col  = jt * 16 + lr;

    const float* wi0 = w_ih + (size_t)(col)        * 1024 + 2 * half;
    const float* wi1 = w_ih + (size_t)(col + 512)  * 1024 + 2 * half;
    const float* wi2 = w_ih + (size_t)(col + 1024) * 1024 + 2 * half;
    const float* wh0 = w_hh + (size_t)(col)        * 512 + 2 * half;
    const float* wh1 = w_hh + (size_t)(col + 512)  * 512 + 2 * half;
    const float* wh2 = w_hh + (size_t)(col + 1024) * 512 + 2 * half;
    __builtin_prefetch(wi0, 0, 0);
    __builtin_prefetch(wi1, 0, 0);
    __builtin_prefetch(wi2, 0, 0);
    __builtin_prefetch(wh0, 0, 0);
    __builtin_prefetch(wh1, 0, 0);
    __builtin_prefetch(wh2, 0, 0);

    v8f air = {0,0,0,0,0,0,0,0}, aiz = {0,0,0,0,0,0,0,0}, ain = {0,0,0,0,0,0,0,0};
    v8f ahr = {0,0,0,0,0,0,0,0}, ahz = {0,0,0,0,0,0,0,0}, ahn = {0,0,0,0,0,0,0,0};

    // gi phase 1: k in [0,512) over r
    {
        const float* ap = r + (size_t)arow * 512 + 2 * half;
        for (int k0 = 0; k0 < 512; k0 += 4) {
            const v2f a = ld2(ap); ap += 4;
            const v2f b0 = ld2(wi0); wi0 += 4;
            const v2f b1 = ld2(wi1); wi1 += 4;
            const v2f b2 = ld2(wi2); wi2 += 4;
            air = wmma_f32_4(a, b0, air);
            aiz = wmma_f32_4(a, b1, aiz);
            ain = wmma_f32_4(a, b2, ain);
        }
    }
    // gi phase 2: k in [512,1024) over g (wi pointers continue)
    {
        const float* ap = g + (size_t)arow * 512 + 2 * half;
        for (int k0 = 0; k0 < 512; k0 += 4) {
            const v2f a = ld2(ap); ap += 4;
            const v2f b0 = ld2(wi0); wi0 += 4;
            const v2f b1 = ld2(wi1); wi1 += 4;
            const v2f b2 = ld2(wi2); wi2 += 4;
            air = wmma_f32_4(a, b0, air);
            aiz = wmma_f32_4(a, b1, aiz);
            ain = wmma_f32_4(a, b2, ain);
        }
    }
    // gh: k in [0,512) over s
    {
        const float* ap = s + (size_t)arow * 512 + 2 * half;
        for (int k0 = 0; k0 < 512; k0 += 4) {
            const v2f a = ld2(ap); ap += 4;
            const v2f b0 = ld2(wh0); wh0 += 4;
            const v2f b1 = ld2(wh1); wh1 += 4;
            const v2f b2 = ld2(wh2); wh2 += 4;
            ahr = wmma_f32_4(a, b0, ahr);
            ahz = wmma_f32_4(a, b1, ahz);
            ahn = wmma_f32_4(a, b2, ahn);
        }
    }

    const float bi0 = b_ih[col], bi1 = b_ih[col + 512], bi2 = b_ih[col + 1024];
    const float bh0 = b_hh[col], bh1 = b_hh[col + 512], bh2 = b_hh[col + 1024];

#pragma unroll
    for (int i = 0; i < 8; ++i) {
        const int row = mt * 16 + i + 8 * half;
        const float ir  = air[i] + bi0;
        const float iz  = aiz[i] + bi1;
        const float inn = ain[i] + bi2;
        const float hr  = ahr[i] + bh0;
        const float hz  = ahz[i] + bh1;
        const float hn  = ahn[i] + bh2;
        const float rg = sigmoidf_(ir + hr);
        const float zg = sigmoidf_(iz + hz);
        const float ng = tanhf(inn + rg * hn);
        const float sold = s[(size_t)row * 512 + col];
        s_out[(size_t)row * 512 + col] = (1.0f - zg) * ng + zg * sold;
    }
}

// ---------------------------------------------------------------------------
// Attention: scores = q . k_tok, masked softmax over N=64, r = alpha . v_tok.
// One block per batch element, 64 threads (2 waves).
// ---------------------------------------------------------------------------
__global__ void __launch_bounds__(64)
attn_kernel(const float* __restrict__ q, const float* __restrict__ ktok,
            const float* __restrict__ vtok, const int* __restrict__ mask,
            float* __restrict__ r)
{
    const int b = blockIdx.x;
    const int t = threadIdx.x;

    __shared__ float qs[DATT];
    __shared__ float sc[NN];
    __shared__ float ex[NN];

    for (int d = t; d < DATT; d += 64) qs[d] = q[(size_t)b * DATT + d];
    __syncthreads();

    {
        const float* kp = ktok + ((size_t)b * NN + t) * DATT;
        float acc = 0.f;
        for (int d = 0; d < DATT; ++d) acc += qs[d] * kp[d];
        if (mask[b * NN + t] == 0) acc = -1e9f;
        sc[t] = acc;
    }
    __syncthreads();

    float mx = -INFINITY;
    for (int n = 0; n < NN; ++n) mx = fmaxf(mx, sc[n]);
    ex[t] = expf(sc[t] - mx);
    __syncthreads();

    float sum = 0.f;
    for (int n = 0; n < NN; ++n) sum += ex[n];
    const float inv = 1.0f / sum;

    for (int d = t; d < DATT; d += 64) {
        float acc = 0.f;
        for (int n = 0; n < NN; ++n)
            acc += ex[n] * vtok[((size_t)b * NN + n) * DATT + d];
        r[(size_t)b * DATT + d] = acc * inv;
    }
}

// ---------------------------------------------------------------------------
// loc GEMV (2 of 3 outputs used) + patch scatter-add into padded canvas.
// One block per batch element, 512 threads.
// ---------------------------------------------------------------------------
__global__ void __launch_bounds__(512)
loc_scatter_kernel(const float* __restrict__ s, const float* __restrict__ r,
                   const float* __restrict__ g,
                   const float* __restrict__ w_loc, const float* __restrict__ b_loc,
                   const float* __restrict__ patch, float* __restrict__ canvas)
{
    const int b = blockIdx.x;
    const int t = threadIdx.x;

    __shared__ float red0[512], red1[512];
    __shared__ int ij[2];

    float p0 = 0.f, p1 = 0.f;
    for (int k = t; k < 1536; k += 512) {
        float z;
        if (k < 512)        z = s[(size_t)b * 512 + k];
        else if (k < 1024)  z = r[(size_t)b * 512 + (k - 512)];
        else                z = g[(size_t)b * 512 + (k - 1024)];
        p0 += z * w_loc[k];
        p1 += z * w_loc[1536 + k];
    }
    red0[t] = p0; red1[t] = p1;
    __syncthreads();
    for (int off = 256; off > 0; off >>= 1) {
        if (t < off) { red0[t] += red0[t + off]; red1[t] += red1[t + off]; }
        __syncthreads();
    }
    if (t == 0) {
        const float cx = tanhf(red0[0] + b_loc[0]);
        const float cy = tanhf(red1[0] + b_loc[1]);
        int j = (int)rintf(0.5f * (cx + 1.0f) * (float)(WW - 1));
        int i = (int)rintf(0.5f * (cy + 1.0f) * (float)(HH - 1));
        j = min(max(j, 0), WW - 1);
        i = min(max(i, 0), HH - 1);
        ij[0] = i; ij[1] = j;
    }
    __syncthreads();

    if (t < CC * KK * KK) {  // 400
        const int i = ij[0], j = ij[1];
        const int c  = t / (KK * KK);
        const int rm = t % (KK * KK);
        const int ki = rm / KK;
        const int kj = rm % KK;
        const float val = patch[(size_t)b * (CC * KK * KK) + t];
        const size_t idx = (((size_t)b * HP + (i + ki)) * WP + (j + kj)) * CC + c;
        canvas[idx] += val;   // unique index per thread within this launch
    }
}

// ---------------------------------------------------------------------------
// Crop padded NHWC canvas -> NCHW output.
// ---------------------------------------------------------------------------
__global__ void __launch_bounds__(256)
crop_kernel(const float* __restrict__ canvas, float* __restrict__ out)
{
    const size_t total = (size_t)BB * CC * HH * WW;
    const size_t idx = (size_t)blockIdx.x * 256 + threadIdx.x;
    if (idx >= total) return;
    const int w = (int)(idx & (WW - 1));
    const size_t r1 = idx >> 7;
    const int h = (int)(r1 & (HH - 1));
    const size_t r2 = r1 >> 7;
    const int c = (int)(r2 & (CC - 1));
    const int b = (int)(r2 >> 4);
    out[idx] = canvas[(((size_t)b * HP + (h + HALF)) * WP + (w + HALF)) * CC + c];
}

// ---------------------------------------------------------------------------
// Host orchestration
// ---------------------------------------------------------------------------
extern "C" void kernel_launch(void* const* d_in, const int* in_sizes, int n_in,
                              void* d_out, int out_size, void* d_ws, size_t ws_size,
                              hipStream_t stream) {
    const float* token   = (const float*)d_in[0];   // [64,64,512]
    const float* grep    = (const float*)d_in[1];   // [64,512]
    const int*   mask    = (const int*)  d_in[2];   // [64,64]
    const float* w_init  = (const float*)d_in[3];
    const float* b_init  = (const float*)d_in[4];
    const float* wq      = (const float*)d_in[5];
    const float* bq      = (const float*)d_in[6];
    const float* wk      = (const float*)d_in[7];
    const float* bk      = (const float*)d_in[8];
    const float* wv      = (const float*)d_in[9];
    const float* bv      = (const float*)d_in[10];
    const float* w_ih    = (const float*)d_in[11];
    const float* b_ih    = (const float*)d_in[12];
    const float* w_hh    = (const float*)d_in[13];
    const float* b_hh    = (const float*)d_in[14];
    const float* w_write = (const float*)d_in[15];
    const float* b_write = (const float*)d_in[16];
    const float* w_loc   = (const float*)d_in[17];
    const float* b_loc   = (const float*)d_in[18];

    float* ws = (float*)d_ws;
    size_t off = 0;
    float* canvas = ws + off; off += (size_t)BB * HP * WP * CC;   // 17.84M floats
    float* ktok   = ws + off; off += (size_t)BB * NN * DATT;      // 2.10M
    float* vtok   = ws + off; off += (size_t)BB * NN * DATT;      // 2.10M
    float* sA     = ws + off; off += (size_t)BB * DSTATE;
    float* sB     = ws + off; off += (size_t)BB * DSTATE;
    float* qb     = ws + off; off += (size_t)BB * DATT;
    float* rb     = ws + off; off += (size_t)BB * DATT;
    float* patch  = ws + off; off += (size_t)BB * CC * KK * KK;

    hipMemsetAsync(canvas, 0, (size_t)BB * HP * WP * CC * sizeof(float), stream);

    const dim3 wblk(32);

    // N=512 GEMMs: 32 N-tiles -> 8 groups of NB=4
    // k_tok = token @ wk^T + bk   (M=4096)
    wmma_gemm_xt<4><<<dim3(256 * 8), wblk, 0, stream>>>(
        token, 512, nullptr, 0, wk, bk, ktok, 512, /*act=*/0, /*ngroupsN=*/8);
    // v_tok = token @ wv^T + bv
    wmma_gemm_xt<4><<<dim3(256 * 8), wblk, 0, stream>>>(
        token, 512, nullptr, 0, wv, bv, vtok, 512, 0, 8);
    // s0 = tanh(global @ w_init^T + b_init)  (M=64)
    wmma_gemm_xt<4><<<dim3(4 * 8), wblk, 0, stream>>>(
        grep, 512, nullptr, 0, w_init, b_init, sA, 512, /*act=*/1, 8);

    for (int t = 0; t < TT; ++t) {
        float* scur  = (t & 1) ? sB : sA;
        float* snext = (t & 1) ? sA : sB;

        // q = s @ wq^T + bq
        wmma_gemm_xt<4><<<dim3(4 * 8), wblk, 0, stream>>>(
            scur, 512, nullptr, 0, wq, bq, qb, 512, 0, 8);
        // attention -> r
        attn_kernel<<<dim3(BB), dim3(64), 0, stream>>>(qb, ktok, vtok, mask, rb);
        // GRU -> s_next
        gru_kernel<<<dim3(128), wblk, 0, stream>>>(
            rb, grep, scur, w_ih, b_ih, w_hh, b_hh, snext);
        // patch = [s_next | r] @ w_write^T + b_write  (M=64, N=400: 25 tiles -> 5 groups of NB=5)
        wmma_gemm_xt<5><<<dim3(4 * 5), wblk, 0, stream>>>(
            snext, 512, rb, 512, w_write, b_write, patch, 400, 0, 5);
        // loc + scatter-add into padded canvas
        loc_scatter_kernel<<<dim3(BB), dim3(512), 0, stream>>>(
            snext, rb, grep, w_loc, b_loc, patch, canvas);
    }

    const size_t total = (size_t)BB * CC * HH * WW;
    crop_kernel<<<dim3((unsigned)((total + 255) / 256)), dim3(256), 0, stream>>>(
        canvas, (float*)d_out);
}